// DGCNN_7473243095568
// MI455X (gfx1250) — compile-verified
//
#include <hip/hip_runtime.h>
#include <hip/hip_bf16.h>

#define NPTS 2048
#define KNB  20
#define NB   4          // batch
#define NEG_SLOPE 0.2f

typedef float v2f __attribute__((ext_vector_type(2)));
typedef float v8f __attribute__((ext_vector_type(8)));

// ---------------------------------------------------------------------------
// fp32 WMMA: D(16x16) += A(16x4) * B(4x16), one wave32.
// A layout (ISA 7.12.2): lanes 0-15 hold M=lane, K={0,1}; lanes 16-31 K={2,3}.
// B layout mirrors: lane holds N=lane&15, K slots {klo, klo+1}, klo=(lane>>4)*2.
// C/D layout: VGPR r -> M = r + (lane>>4)*8, N = lane&15.
// ---------------------------------------------------------------------------
__device__ inline v8f wmma4(v2f a, v2f b, v8f c) {
    return __builtin_amdgcn_wmma_f32_16x16x4_f32(false, a, false, b,
                                                 (short)0, c, false, false);
}

__device__ inline void atomicMaxFloat(float* addr, float val) {
    int iv = __float_as_int(val);
    if (iv >= 0) atomicMax((int*)addr, iv);
    else         atomicMin((unsigned int*)addr, (unsigned int)iv);
}

// ------------------------- small utility kernels ---------------------------

__global__ void copy_k(const float* __restrict__ s, float* __restrict__ d, int n) {
    int t = blockIdx.x * blockDim.x + threadIdx.x;
    if (t < n) d[t] = s[t];
}

__global__ void zero_k(float* p, int n) {
    int t = blockIdx.x * blockDim.x + threadIdx.x;
    if (t < n) p[t] = 0.f;
}

// pad x (B,3,N) -> xpad (B,4,N), channel 3 = 0
__global__ void pad_x(const float* __restrict__ x, float* __restrict__ xpad) {
    int t = blockIdx.x * blockDim.x + threadIdx.x;
    if (t >= NB * 4 * NPTS) return;
    int b = t / (4 * NPTS);
    int r = t - b * 4 * NPTS;
    int c = r / NPTS, n = r & (NPTS - 1);
    xpad[t] = (c < 3) ? x[b * 3 * NPTS + c * NPTS + n] : 0.f;
}

// pad w1 (64,6) -> (64,8): cols [n0,n1,n2,0, c0,c1,c2,0]
__global__ void pad_w1(const float* __restrict__ w, float* __restrict__ wp) {
    int t = blockIdx.x * blockDim.x + threadIdx.x;
    if (t >= 64 * 8) return;
    int o = t >> 3, k = t & 7;
    float v = 0.f;
    if (k < 3)                 v = w[o * 6 + k];
    else if (k >= 4 && k < 7)  v = w[o * 6 + k - 1];
    wp[t] = v;
}

// fill one channel-slice [coff, coff+Cout) of xcat (B,512,N) with -inf
__global__ void fill_ninf(float* xcat, int coff, int Cout) {
    int t = blockIdx.x * blockDim.x + threadIdx.x;
    int tot = NB * Cout * NPTS;
    if (t >= tot) return;
    int b = t / (Cout * NPTS);
    int r = t - b * (Cout * NPTS);
    xcat[b * 512 * NPTS + coff * NPTS + r] = __int_as_float(0xFF800000);
}

__global__ void norms_k(const float* __restrict__ x, float* __restrict__ norms,
                        int Cin, int bs) {
    int t = blockIdx.x * blockDim.x + threadIdx.x;
    if (t >= NB * NPTS) return;
    int b = t / NPTS, n = t & (NPTS - 1);
    const float* xb = x + b * bs + n;
    float s = 0.f;
    for (int c = 0; c < Cin; ++c) { float v = xb[c * NPTS]; s += v * v; }
    norms[t] = s;
}

__global__ void finalize_bn(const float* __restrict__ sums, const float* __restrict__ sumsq,
                            const float* __restrict__ g, const float* __restrict__ bt,
                            float* __restrict__ scale, float* __restrict__ bias,
                            int C, float inv_count) {
    int c = blockIdx.x * blockDim.x + threadIdx.x;
    if (c >= C) return;
    float m = sums[c] * inv_count;
    float v = sumsq[c] * inv_count - m * m;
    float s = g[c] * rsqrtf(v + 1e-5f);
    scale[c] = s;
    bias[c]  = bt[c] - m * s;
}

// ------------------------ kNN: fused WMMA Gram + top-k ---------------------
// One wave: 32 query rows x 32 candidate cols per sweep step (2x2 WMMA tiles,
// A and B operands each reused twice). All 32 lanes keep one sorted top-20
// list each (desc, lowest index first on ties, matching lax.top_k).

template<int CIN>
__global__ void __launch_bounds__(32)
knn_topk(const float* __restrict__ x, const float* __restrict__ norms,
         int* __restrict__ idx, int bs) {
    __shared__ float tile[32 * 32];
    __shared__ float sh_xxj[32];
    __shared__ float lv[32 * KNB];
    __shared__ int   li[32 * KNB];

    int lane = threadIdx.x;
    int b    = blockIdx.y;
    int i0   = blockIdx.x * 32;
    const float* xb  = x + b * bs;
    const float* nrm = norms + b * NPTS;

    int m    = lane & 15;
    int klo  = (lane >> 4) * 2;
    int half = lane >> 4;
    float xxi = nrm[i0 + lane];
    int cnt = 0;

    for (int jt = 0; jt < NPTS / 32; ++jt) {
        int j0 = jt * 32;
        v8f c00 = {}, c01 = {}, c10 = {}, c11 = {};
#pragma unroll 4
        for (int cc = 0; cc < CIN; cc += 4) {
            int k0 = cc + klo;
            const float* r0 = xb + k0 * NPTS;
            const float* r1 = xb + (k0 + 1) * NPTS;
            v2f aA, aB, b0, b1;
            aA.x = r0[i0 + m];       aA.y = r1[i0 + m];
            aB.x = r0[i0 + 16 + m];  aB.y = r1[i0 + 16 + m];
            b0.x = r0[j0 + m];       b0.y = r1[j0 + m];
            b1.x = r0[j0 + 16 + m];  b1.y = r1[j0 + 16 + m];
            c00 = wmma4(aA, b0, c00);
            c01 = wmma4(aA, b1, c01);
            c10 = wmma4(aB, b0, c10);
            c11 = wmma4(aB, b1, c11);
        }
        sh_xxj[lane] = nrm[j0 + lane];
        for (int r = 0; r < 8; ++r) {
            int row = r + half * 8;
            tile[row * 32 + m]             = c00[r];
            tile[row * 32 + 16 + m]        = c01[r];
            tile[(16 + row) * 32 + m]      = c10[r];
            tile[(16 + row) * 32 + 16 + m] = c11[r];
        }
        __syncthreads();
        {
            float* v  = lv + lane * KNB;
            int*   id = li + lane * KNB;
            for (int jj = 0; jj < 32; ++jj) {
                float val = 2.f * tile[lane * 32 + jj] - xxi - sh_xxj[jj];
                int colj = j0 + jj;
                if (cnt < KNB) {
                    int p = cnt++;
                    while (p > 0 && v[p - 1] < val) { v[p] = v[p - 1]; id[p] = id[p - 1]; --p; }
                    v[p] = val; id[p] = colj;
                } else if (val > v[KNB - 1]) {
                    int p = KNB - 1;
                    while (p > 0 && v[p - 1] < val) { v[p] = v[p - 1]; id[p] = id[p - 1]; --p; }
                    v[p] = val; id[p] = colj;
                }
            }
        }
        __syncthreads();
    }
    {
        int* id  = li + lane * KNB;
        int* dst = idx + b * NPTS * KNB + (i0 + lane) * KNB;
        for (int q = 0; q < KNB; ++q) dst[q] = id[q];
    }
}

// ------------------- EdgeConv GEMM, 2x2 tiles per wave ---------------------
// out[co, nk] = sum_ci w[co,ci] * F[ci,nk];  F = [x[.,idx] ; x[.,center]]
// Branchless neighbor/center address select; 64-bit weight loads; A and B
// operand fetches each feed two WMMAs.

template<int CIN>
__device__ inline void edge_conv_tile4(const float* __restrict__ xb,
                                       const int* __restrict__ idxb,
                                       const float* __restrict__ w,
                                       int co0, int nk0, int lane,
                                       v8f& c00, v8f& c01, v8f& c10, v8f& c11) {
    const int CIN2 = 2 * CIN;
    int col = lane & 15;
    int nkA = nk0 + col;
    int npA = nkA / KNB, kkA = nkA - npA * KNB;
    int jA  = idxb[npA * KNB + kkA];
    int nkB = nk0 + 16 + col;
    int npB = nkB / KNB, kkB = nkB - npB * KNB;
    int jB  = idxb[npB * KNB + kkB];

    int m   = lane & 15;
    int klo = (lane >> 4) * 2;
    const v2f* wrowA = (const v2f*)(w + (co0 + m) * CIN2);
    const v2f* wrowB = (const v2f*)(w + (co0 + 16 + m) * CIN2);

    c00 = {}; c01 = {}; c10 = {}; c11 = {};
#pragma unroll 4
    for (int ci0 = 0; ci0 < CIN2; ci0 += 4) {
        int k0 = ci0 + klo, k1 = k0 + 1;
        v2f a0 = wrowA[k0 >> 1];
        v2f a1 = wrowB[k0 >> 1];
        // neighbor channels [0,CIN), center channels [CIN,2CIN)
        int ch0  = (k0 < CIN) ? k0 : k0 - CIN;
        int pt0A = (k0 < CIN) ? jA : npA;
        int pt0B = (k0 < CIN) ? jB : npB;
        int ch1  = (k1 < CIN) ? k1 : k1 - CIN;
        int pt1A = (k1 < CIN) ? jA : npA;
        int pt1B = (k1 < CIN) ? jB : npB;
        v2f b0, b1;
        b0.x = xb[ch0 * NPTS + pt0A];
        b0.y = xb[ch1 * NPTS + pt1A];
        b1.x = xb[ch0 * NPTS + pt0B];
        b1.y = xb[ch1 * NPTS + pt1B];
        c00 = wmma4(a0, b0, c00);
        c01 = wmma4(a0, b1, c01);
        c10 = wmma4(a1, b0, c10);
        c11 = wmma4(a1, b1, c11);
    }
}

// 32x16 LDS reduction helper: rows co0..co0+31, each lane reduces one row.
__device__ inline void reduce_pair(float* tile, const v8f& ca, const v8f& cb,
                                   int lane, float& s, float& s2) {
    int m = lane & 15, half = lane >> 4;
    for (int r = 0; r < 8; ++r) {
        int row = r + half * 8;
        tile[row * 16 + m]        = ca[r];
        tile[(16 + row) * 16 + m] = cb[r];
    }
    __syncthreads();
    for (int q = 0; q < 16; ++q) { float v = tile[lane * 16 + q]; s += v; s2 += v * v; }
    __syncthreads();
}

// Pass A: conv tiles -> per-channel sum / sumsq (BN training stats)
template<int CIN>
__global__ void __launch_bounds__(32)
edge_stats(const float* __restrict__ x, const int* __restrict__ idx,
           const float* __restrict__ w, float* __restrict__ sums,
           float* __restrict__ sumsq, int bs) {
    __shared__ float tile[32 * 16];
    int lane = threadIdx.x;
    int b = blockIdx.z;
    v8f c00, c01, c10, c11;
    edge_conv_tile4<CIN>(x + b * bs, idx + b * NPTS * KNB, w,
                         blockIdx.x * 32, blockIdx.y * 32, lane, c00, c01, c10, c11);
    float s = 0.f, s2 = 0.f;
    reduce_pair(tile, c00, c10, lane, s, s2);
    reduce_pair(tile, c01, c11, lane, s, s2);
    atomicAdd(&sums[blockIdx.x * 32 + lane], s);
    atomicAdd(&sumsq[blockIdx.x * 32 + lane], s2);
}

// Pass B: recompute conv tiles, BN + LeakyReLU, max-over-k via atomic fmax
template<int CIN>
__global__ void __launch_bounds__(32)
edge_apply(const float* __restrict__ x, const int* __restrict__ idx,
           const float* __restrict__ w, const float* __restrict__ scale,
           const float* __restrict__ bias, float* __restrict__ out,
           int bs, int bso) {
    int lane = threadIdx.x;
    int b = blockIdx.z;
    int co0 = blockIdx.x * 32, nk0 = blockIdx.y * 32;
    v8f c00, c01, c10, c11;
    edge_conv_tile4<CIN>(x + b * bs, idx + b * NPTS * KNB, w, co0, nk0, lane,
                         c00, c01, c10, c11);
    int half = lane >> 4, col = lane & 15;
    int nkA = nk0 + col,      npA = nkA / KNB;
    int nkB = nk0 + 16 + col, npB = nkB / KNB;
    float* ob = out + b * bso;
    for (int r = 0; r < 8; ++r) {
        int coA = co0 + r + half * 8;
        int coB = coA + 16;
        float scA = scale[coA], biA = bias[coA];
        float scB = scale[coB], biB = bias[coB];
        float y;
        y = c00[r] * scA + biA; y = y > 0.f ? y : NEG_SLOPE * y;
        atomicMaxFloat(&ob[coA * NPTS + npA], y);
        y = c01[r] * scA + biA; y = y > 0.f ? y : NEG_SLOPE * y;
        atomicMaxFloat(&ob[coA * NPTS + npB], y);
        y = c10[r] * scB + biB; y = y > 0.f ? y : NEG_SLOPE * y;
        atomicMaxFloat(&ob[coB * NPTS + npA], y);
        y = c11[r] * scB + biB; y = y > 0.f ? y : NEG_SLOPE * y;
        atomicMaxFloat(&ob[coB * NPTS + npB], y);
    }
}

// --------------- final 512x512 pointwise conv, 2x2 tiles -------------------

__device__ inline void gemm512_tile4(const float* __restrict__ xcat,
                                     const float* __restrict__ w,
                                     int co0, int g0, int lane,
                                     v8f& c00, v8f& c01, v8f& c10, v8f& c11) {
    int col = lane & 15;
    int gA = g0 + col,      bA = gA >> 11, nA = gA & (NPTS - 1);
    int gB = g0 + 16 + col, bB = gB >> 11, nB = gB & (NPTS - 1);
    const float* xbA = xcat + bA * 512 * NPTS;
    const float* xbB = xcat + bB * 512 * NPTS;
    int m   = lane & 15;
    int klo = (lane >> 4) * 2;
    const v2f* wrowA = (const v2f*)(w + (co0 + m) * 512);
    const v2f* wrowB = (const v2f*)(w + (co0 + 16 + m) * 512);
    c00 = {}; c01 = {}; c10 = {}; c11 = {};
#pragma unroll 4
    for (int ci0 = 0; ci0 < 512; ci0 += 4) {
        int k0 = ci0 + klo;
        v2f a0 = wrowA[k0 >> 1];
        v2f a1 = wrowB[k0 >> 1];
        v2f b0, b1;
        b0.x = xbA[k0 * NPTS + nA];
        b0.y = xbA[(k0 + 1) * NPTS + nA];
        b1.x = xbB[k0 * NPTS + nB];
        b1.y = xbB[(k0 + 1) * NPTS + nB];
        c00 = wmma4(a0, b0, c00);
        c01 = wmma4(a0, b1, c01);
        c10 = wmma4(a1, b0, c10);
        c11 = wmma4(a1, b1, c11);
    }
}

__global__ void __launch_bounds__(32)
final_stats(const float* __restrict__ xcat, const float* __restrict__ w,
            float* __restrict__ sums, float* __restrict__ sumsq) {
    __shared__ float tile[32 * 16];
    int lane = threadIdx.x;
    v8f c00, c01, c10, c11;
    gemm512_tile4(xcat, w, blockIdx.x * 32, blockIdx.y * 32, lane, c00, c01, c10, c11);
    float s = 0.f, s2 = 0.f;
    reduce_pair(tile, c00, c10, lane, s, s2);
    reduce_pair(tile, c01, c11, lane, s, s2);
    atomicAdd(&sums[blockIdx.x * 32 + lane], s);
    atomicAdd(&sumsq[blockIdx.x * 32 + lane], s2);
}

__global__ void __launch_bounds__(32)
final_apply(const float* __restrict__ xcat, const float* __restrict__ w,
            const float* __restrict__ scale, const float* __restrict__ bias,
            float* __restrict__ out) {
    int lane = threadIdx.x;
    int co0 = blockIdx.x * 32, g0 = blockIdx.y * 32;
    v8f c00, c01, c10, c11;
    gemm512_tile4(xcat, w, co0, g0, lane, c00, c01, c10, c11);
    int half = lane >> 4, col = lane & 15;
    int gA = g0 + col,      bA = gA >> 11, nA = gA & (NPTS - 1);
    int gB = g0 + 16 + col, bB = gB >> 11, nB = gB & (NPTS - 1);
    for (int r = 0; r < 8; ++r) {
        int coA = co0 + r + half * 8;
        int coB = coA + 16;
        float scA = scale[coA], biA = bias[coA];
        float scB = scale[coB], biB = bias[coB];
        float y;
        y = c00[r] * scA + biA; y = y > 0.f ? y : NEG_SLOPE * y;
        out[bA * 512 * NPTS + coA * NPTS + nA] = y;
        y = c01[r] * scA + biA; y = y > 0.f ? y : NEG_SLOPE * y;
        out[bB * 512 * NPTS + coA * NPTS + nB] = y;
        y = c10[r] * scB + biB; y = y > 0.f ? y : NEG_SLOPE * y;
        out[bA * 512 * NPTS + coB * NPTS + nA] = y;
        y = c11[r] * scB + biB; y = y > 0.f ? y : NEG_SLOPE * y;
        out[bB * 512 * NPTS + coB * NPTS + nB] = y;
    }
}

// ------------------------------- orchestration -----------------------------

template<int CIN>
static void run_block(const float* in, int bs, const float* w, const float* g,
                      const float* bt, int Cout, int coff,
                      float* xcat, float* norms, int* idx,
                      float* sums, float* sumsq, float* scale, float* bias,
                      hipStream_t stream) {
    norms_k<<<(NB * NPTS + 255) / 256, 256, 0, stream>>>(in, norms, CIN, bs);
    knn_topk<CIN><<<dim3(NPTS / 32, NB), 32, 0, stream>>>(in, norms, idx, bs);

    zero_k<<<4, 256, 0, stream>>>(sums, 1024);  // sums + sumsq contiguous
    edge_stats<CIN><<<dim3(Cout / 32, NPTS * KNB / 32, NB), 32, 0, stream>>>(
        in, idx, w, sums, sumsq, bs);
    finalize_bn<<<(Cout + 63) / 64, 64, 0, stream>>>(
        sums, sumsq, g, bt, scale, bias, Cout, 1.f / (float)(NB * NPTS * KNB));

    fill_ninf<<<(NB * Cout * NPTS + 255) / 256, 256, 0, stream>>>(xcat, coff, Cout);
    edge_apply<CIN><<<dim3(Cout / 32, NPTS * KNB / 32, NB), 32, 0, stream>>>(
        in, idx, w, scale, bias, xcat + coff * NPTS, bs, 512 * NPTS);
}

extern "C" void kernel_launch(void* const* d_in, const int* in_sizes, int n_in,
                              void* d_out, int out_size, void* d_ws, size_t ws_size,
                              hipStream_t stream) {
    (void)in_sizes; (void)n_in; (void)out_size; (void)ws_size;
    const float* x = (const float*)d_in[0];
    const float *w[5], *g[5], *bt[5];
    for (int i = 0; i < 5; ++i) {
        w[i]  = (const float*)d_in[1 + 3 * i];
        g[i]  = (const float*)d_in[2 + 3 * i];
        bt[i] = (const float*)d_in[3 + 3 * i];
    }

    float* ws    = (float*)d_ws;
    float* norms = ws;                        // NB*NPTS      (8192)
    float* sums  = ws + 8192;                 // 512
    float* sumsq = ws + 8704;                 // 512
    float* scale = ws + 9216;                 // 512
    float* bias  = ws + 9728;                 // 512
    float* xcat  = ws + 10240;                // NB*512*NPTS  (4,194,304)
    int*   idx   = (int*)(xcat + NB * 512 * NPTS);   // NB*NPTS*KNB ints (163,840)
    float* xpad  = (float*)(idx + NB * NPTS * KNB);  // NB*4*NPTS (32,768)
    float* wpad  = xpad + NB * 4 * NPTS;             // 64*8 (512)

    float* out = (float*)d_out;
    // tuple output 0: passthrough of x (B,3,N)
    copy_k<<<(NB * 3 * NPTS + 255) / 256, 256, 0, stream>>>(x, out, NB * 3 * NPTS);
    float* out5 = out + NB * 3 * NPTS;

    // zero-pad block 1 so every GEMM K-loop is guard-free (Cin2 % 4 == 0)
    pad_x<<<(NB * 4 * NPTS + 255) / 256, 256, 0, stream>>>(x, xpad);
    pad_w1<<<2, 256, 0, stream>>>(w[0], wpad);

    run_block<4>  (xpad,               4 * NPTS,   wpad, g[0], bt[0],  64,   0,
                   xcat, norms, idx, sums, sumsq, scale, bias, stream);
    run_block<64> (xcat,               512 * NPTS, w[1], g[1], bt[1],  64,  64,
                   xcat, norms, idx, sums, sumsq, scale, bias, stream);
    run_block<64> (xcat +  64 * NPTS,  512 * NPTS, w[2], g[2], bt[2], 128, 128,
                   xcat, norms, idx, sums, sumsq, scale, bias, stream);
    run_block<128>(xcat + 128 * NPTS,  512 * NPTS, w[3], g[3], bt[3], 256, 256,
                   xcat, norms, idx, sums, sumsq, scale, bias, stream);

    // final conv over the concatenated features (B,512,N)
    zero_k<<<4, 256, 0, stream>>>(sums, 1024);
    final_stats<<<dim3(512 / 32, NB * NPTS / 32), 32, 0, stream>>>(xcat, w[4], sums, sumsq);
    finalize_bn<<<8, 64, 0, stream>>>(sums, sumsq, g[4], bt[4], scale, bias, 512,
                                      1.f / (float)(NB * NPTS));
    final_apply<<<dim3(512 / 32, NB * NPTS / 32), 32, 0, stream>>>(xcat, w[4], scale, bias, out5);
}